// GAE_24601572671537
// MI455X (gfx1250) — compile-verified
//
#include <hip/hip_runtime.h>

// ---------------------------------------------------------------------------
// GAE backward scan, CDNA5 (gfx1250), wave32.
// Bandwidth-bound (~384 MiB @ 23.3 TB/s ≈ 17 us). Strategy:
//   - 1 lane per (batch row, tail) sequence; 32 lanes = 4 rows x 8 tails.
//   - Stage 16-timestep chunks of {reward, terminated, value, next_value}
//     into LDS via GLOBAL_LOAD_ASYNC_TO_LDS_B128 (coalesced 512B per wave op),
//     double-buffered, synchronized with s_wait_asynccnt.
//   - LDS row stride 544B (== 8 dwords mod 64 banks) -> conflict-free scan.
//   - Outputs written directly (32B runs per (row,t); adjacent t merge in GL2).
// ---------------------------------------------------------------------------

#define B_DIM 2048
#define T_DIM 1024
#define TAIL  8

#define ROWS    4                    // batch rows per block -> 32 threads = 1 wave
#define CH      16                   // timesteps per LDS chunk
#define NCHUNK  (T_DIM / CH)         // 64
#define THREADS (ROWS * TAIL)        // 32

#define ROW_BYTES  (CH * TAIL * 4)        // 512 data bytes per row per chunk
#define ROW_STRIDE (ROW_BYTES + 32)       // 544 -> bank shift of 8 per row
#define ARR_BYTES  (ROWS * ROW_STRIDE)    // 2176
#define SLOT_BYTES (4 * ARR_BYTES)        // 8704 (4 staged arrays)
#define NSLOT      2                      // double buffer -> 17408 B LDS

#define GAMMA_F 0.99f
#define GL_F    (0.99f * 0.95f)

// Exact pointee type from clang's diagnostic: int __attribute__((vector_size(16)))
typedef int v4i __attribute__((vector_size(4 * sizeof(int))));
typedef __attribute__((address_space(1))) v4i gl_v4i;   // global (printed "__device__")
typedef __attribute__((address_space(3))) v4i ld_v4i;   // LDS

// ---- async global->LDS 16B-per-lane copy -----------------------------------
#if __has_builtin(__builtin_amdgcn_global_load_async_to_lds_b128)
#pragma message("CDNA5 path: __builtin_amdgcn_global_load_async_to_lds_b128")
#define HAVE_ASYNC_BUILTIN 1
#else
#pragma message("CDNA5 path: inline-asm global_load_async_to_lds_b128 fallback")
#define HAVE_ASYNC_BUILTIN 0
#endif

__device__ __forceinline__ void cp_async_b128(const void* g, const void* l) {
#if HAVE_ASYNC_BUILTIN
  // Integer round-trip: generic LDS pointer low 32 bits == LDS byte offset.
  gl_v4i* gp = (gl_v4i*)(unsigned long long)g;           // non-const, as1
  ld_v4i* lp = (ld_v4i*)(unsigned int)(unsigned long long)l;
  __builtin_amdgcn_global_load_async_to_lds_b128(gp, lp, 0, 0);
#else
  unsigned long long ga = (unsigned long long)g;
  unsigned           la = (unsigned)(unsigned long long)l;
  asm volatile("global_load_async_to_lds_b128 %0, %1, off"
               :: "v"(la), "v"(ga) : "memory");
#endif
}

#if __has_builtin(__builtin_amdgcn_s_wait_asynccnt)
#pragma message("CDNA5 path: __builtin_amdgcn_s_wait_asynccnt")
#define WAIT_ASYNC(N) __builtin_amdgcn_s_wait_asynccnt(N)
#else
#pragma message("CDNA5 path: inline-asm s_wait_asynccnt fallback")
#define WAIT_ASYNC(N) asm volatile("s_wait_asynccnt %0" :: "n"(N) : "memory")
#endif

// Issue all 16 async B128 copies for one chunk (4 arrays x 4 rows);
// each instruction moves 32 lanes x 16B = 512B, fully coalesced per row.
__device__ __forceinline__ void issue_chunk(
    const unsigned char* rw, const unsigned char* tm,
    const unsigned char* vl, const unsigned char* nv,
    unsigned char* slot, int b0, int c, int tid)
{
  const unsigned rowGBytes = (unsigned)T_DIM * TAIL * 4u;            // 32 KiB / b-row
  const unsigned chunkOff  = (unsigned)c * CH * TAIL * 4u;           // within row
#pragma unroll
  for (int k = 0; k < ROWS; ++k) {
    const unsigned goff = (unsigned)(b0 + k) * rowGBytes + chunkOff + (unsigned)tid * 16u;
    const unsigned loff = (unsigned)k * ROW_STRIDE + (unsigned)tid * 16u;
    cp_async_b128(rw + goff, slot + 0 * ARR_BYTES + loff);
    cp_async_b128(tm + goff, slot + 1 * ARR_BYTES + loff);
    cp_async_b128(vl + goff, slot + 2 * ARR_BYTES + loff);
    cp_async_b128(nv + goff, slot + 3 * ARR_BYTES + loff);
  }
}

__global__ void __launch_bounds__(THREADS) gae_scan_kernel(
    const float* __restrict__ reward,
    const int*   __restrict__ terminated,   // bool promoted to int32 by harness
    const float* __restrict__ value,
    const float* __restrict__ next_value,
    float* __restrict__ adv,
    float* __restrict__ ret)
{
  __shared__ unsigned char smem[NSLOT * SLOT_BYTES];

  const int tid  = threadIdx.x;        // 0..31
  const int b0   = blockIdx.x * ROWS;  // first batch row of this block
  const int r    = tid >> 3;           // 0..3  (row within block)
  const int tail = tid & 7;            // 0..7

  const unsigned char* rwB = (const unsigned char*)reward;
  const unsigned char* tmB = (const unsigned char*)terminated;
  const unsigned char* vlB = (const unsigned char*)value;
  const unsigned char* nvB = (const unsigned char*)next_value;

  // Prologue: prefetch last chunk.
  issue_chunk(rwB, tmB, vlB, nvB, smem + ((NCHUNK - 1) & 1) * SLOT_BYTES,
              b0, NCHUNK - 1, tid);

  const int rowBaseF = r * (ROW_STRIDE / 4);   // 136 floats per row stride
  float gae = 0.0f;

  for (int c = NCHUNK - 1; c >= 0; --c) {
    const int s = c & 1;
    if (c > 0) {
      issue_chunk(rwB, tmB, vlB, nvB, smem + (s ^ 1) * SLOT_BYTES, b0, c - 1, tid);
      WAIT_ASYNC(16);   // chunk c's 16 loads retired (loads complete in order)
    } else {
      WAIT_ASYNC(0);
    }
    __syncthreads();    // single-wave WG: S_NOP in HW, compiler memory fence

    const unsigned char* slot = smem + s * SLOT_BYTES;
    const float* rw = (const float*)(slot + 0 * ARR_BYTES) + rowBaseF;
    const int*   tm = (const int*)  (slot + 1 * ARR_BYTES) + rowBaseF;
    const float* vl = (const float*)(slot + 2 * ARR_BYTES) + rowBaseF;
    const float* nv = (const float*)(slot + 3 * ARR_BYTES) + rowBaseF;

    // Global base for this (row, chunk, tail).
    const long gbase = ((long)(b0 + r) * T_DIM + (long)c * CH) * TAIL + tail;

#pragma unroll
    for (int tt = CH - 1; tt >= 0; --tt) {
      const int li = tt * TAIL + tail;                 // conflict-free banks
      const float v  = vl[li];
      const float nd = (tm[li] != 0) ? 0.0f : 1.0f;
      const float d  = fmaf(GAMMA_F * nd, nv[li], rw[li]) - v;
      gae = fmaf(GL_F * nd, gae, d);
      adv[gbase + (long)tt * TAIL] = gae;
      ret[gbase + (long)tt * TAIL] = gae + v;
    }
    __syncthreads();    // protect slot s^1 reuse ordering (free for 1 wave)
  }
}

extern "C" void kernel_launch(void* const* d_in, const int* in_sizes, int n_in,
                              void* d_out, int out_size, void* d_ws, size_t ws_size,
                              hipStream_t stream) {
  (void)in_sizes; (void)n_in; (void)out_size; (void)d_ws; (void)ws_size;
  const float* reward     = (const float*)d_in[0];
  const int*   terminated = (const int*)  d_in[1];
  const float* value      = (const float*)d_in[2];
  const float* next_value = (const float*)d_in[3];

  float* adv = (float*)d_out;                                  // [B,T,TAIL]
  float* ret = adv + (size_t)B_DIM * T_DIM * TAIL;             // [B,T,TAIL]

  dim3 grid(B_DIM / ROWS);     // 512 single-wave blocks
  dim3 block(THREADS);         // 32 threads = 1 wave32
  gae_scan_kernel<<<grid, block, 0, stream>>>(reward, terminated, value,
                                              next_value, adv, ret);
}